// net_init_68521908240915
// MI455X (gfx1250) — compile-verified
//
#include <hip/hip_runtime.h>
#include <hip/hip_bf16.h>

#define CDIM 144          // GLOBAL_CV
#define NTILE 9           // 144 / 16
#define KSTEPS 36         // 144 / 4   (f32 WMMA K=4)
#define KEDGE 8
#define HIDDENN 3
#define BATCH_G 128

typedef __attribute__((ext_vector_type(2))) float v2f;
typedef __attribute__((ext_vector_type(8))) float v8f;

__device__ __forceinline__ float silu_f(float v) {
    // v * sigmoid(v); v_rcp_f32 instead of IEEE divide expansion
    return v * __builtin_amdgcn_rcpf(1.0f + __expf(-v));
}

// ---------------- degree / norm precompute (graph is static across layers) ----

__global__ __launch_bounds__(256) void deg_init_kernel(float* deg, int n) {
    int t = blockIdx.x * 256 + threadIdx.x;
    if (t < n) deg[t] = 1.0f;                      // self-loop contributes 1
}

__global__ __launch_bounds__(256) void deg_count_kernel(const int* __restrict__ dst,
                                                        float* deg, int e) {
    int t = blockIdx.x * 256 + threadIdx.x;
    if (t < e) unsafeAtomicAdd(&deg[dst[t]], 1.0f);
}

__global__ __launch_bounds__(256) void dinv_kernel(float* deg, int n) {
    int t = blockIdx.x * 256 + threadIdx.x;
    if (t < n) deg[t] = __frsqrt_rn(deg[t]);       // deg >= 1 always
}

__global__ __launch_bounds__(256) void norm_kernel(const int* __restrict__ src,
                                                   const int* __restrict__ dst,
                                                   const float* __restrict__ dinv,
                                                   float* __restrict__ norm, int e) {
    int t = blockIdx.x * 256 + threadIdx.x;
    if (t < e) norm[t] = dinv[src[t]] * dinv[dst[t]];
}

// ---------------- embed: x[n][c] = features[n]*embed_W[c] + embed_b[c] --------

__global__ __launch_bounds__(256) void embed_kernel(const float* __restrict__ feat,
                                                    const float* __restrict__ eW,
                                                    const float* __restrict__ eb,
                                                    float* __restrict__ X, int total) {
    int t = blockIdx.x * 256 + threadIdx.x;
    if (t >= total) return;
    int n = t / CDIM, c = t % CDIM;
    X[t] = feat[n] * eW[c] + eb[c];
}

// ------- fused GCN layer core: L = silu(X) @ W ;  X += bias + dinv^2 * L ------
// block = 256 threads = 8 waves; wave w computes rows [blk*128 + w*16, +16).
// W (144x144 = 81KB) staged in LDS, K-pair swizzled: wl[(k>>1)*288 + n*2 + (k&1)]
// so each lane's B pair (K,K+1 @ col n) is one contiguous 8B ds_load_b64.
// Epilogue applies residual + bias + self-loop term in-register (L stays hot),
// leaving X ready for the edge-scatter atomics.

__global__ __launch_bounds__(256) void gemm_silu_wmma(float* __restrict__ X,
                                                      const float* __restrict__ W,
                                                      float* __restrict__ L,
                                                      const float* __restrict__ dinv,
                                                      const float* __restrict__ bias) {
    __shared__ float wl[72 * 288];                 // 82,944 B (CDNA5: 320KB/WGP)
    const int t = threadIdx.x;
    #pragma unroll
    for (int i = 0; i < 81; ++i) {                 // 81*256 == 144*144
        int idx = t + i * 256;
        int k = idx / CDIM, n = idx % CDIM;
        wl[(k >> 1) * 288 + (n << 1) + (k & 1)] = W[idx];
    }
    __syncthreads();

    const int lane   = t & 31;
    const int wave   = t >> 5;
    const int m_base = blockIdx.x * 128 + wave * 16;
    const int hk     = lane >> 4;                  // 0: K=0,1   1: K=2,3
    const int m_lane = lane & 15;
    const float* xrow = X + (long)(m_base + m_lane) * CDIM + 2 * hk;

    v8f acc[NTILE];
    #pragma unroll
    for (int n = 0; n < NTILE; ++n)
        acc[n] = (v8f){0.f, 0.f, 0.f, 0.f, 0.f, 0.f, 0.f, 0.f};

    for (int ks = 0; ks < KSTEPS; ++ks) {
        const int k4 = ks * 4;
        v2f a;                                     // A 16x4: lane-pair K layout
        a.x = silu_f(xrow[k4]);
        a.y = silu_f(xrow[k4 + 1]);
        const float* bb = wl + (ks * 2 + hk) * 288 + 2 * m_lane;
        #pragma unroll
        for (int n = 0; n < NTILE; ++n) {
            v2f b = *(const v2f*)(bb + n * 32);    // ds_load_b64, bank-clean
            acc[n] = __builtin_amdgcn_wmma_f32_16x16x4_f32(
                false, a, false, b, (short)0, acc[n], false, false);
        }
    }

    // D layout: VGPR r -> M = r + 8*hk, N = lane&15
    const int rbase = m_base + 8 * hk;
    float dv2[8];
    #pragma unroll
    for (int r = 0; r < 8; ++r) {
        float d = dinv[rbase + r];
        dv2[r] = d * d;
    }
    #pragma unroll
    for (int n = 0; n < NTILE; ++n) {
        const int col = n * 16 + m_lane;
        const float bc = bias[col];
        #pragma unroll
        for (int r = 0; r < 8; ++r) {
            long idx = (long)(rbase + r) * CDIM + col;
            float lv = acc[n][r];
            L[idx] = lv;
            X[idx] = X[idx] + bc + dv2[r] * lv;    // residual + bias + self-loop
        }
    }
}

// ---- edge scatter: X[dst] += norm[e] * L[src], one thread per (edge, 4 chans) -

__global__ __launch_bounds__(256) void edge_scatter_kernel(float* __restrict__ X,
                                                           const float* __restrict__ L,
                                                           const int* __restrict__ src,
                                                           const int* __restrict__ dst,
                                                           const float* __restrict__ norm,
                                                           long nwork) {
    long t = (long)blockIdx.x * 256 + threadIdx.x;
    if (t >= nwork) return;
    long e = t / 36;
    int  q = (int)(t - e * 36);
    int  s = src[e];
    int  d = dst[e];
    float nm = norm[e];
    float4 v = *((const float4*)(L + (long)s * CDIM) + q);
    float* xp = X + (long)d * CDIM + q * 4;
    unsafeAtomicAdd(xp + 0, nm * v.x);             // global_atomic_add_f32
    unsafeAtomicAdd(xp + 1, nm * v.y);
    unsafeAtomicAdd(xp + 2, nm * v.z);
    unsafeAtomicAdd(xp + 3, nm * v.w);
}

// ---------------- mean pooling over contiguous 4096-node graphs ---------------

__global__ __launch_bounds__(256) void pool_kernel(const float* __restrict__ X,
                                                   float* __restrict__ P, int npg) {
    int t = blockIdx.x * 256 + threadIdx.x;        // 128*144 threads
    int b = t / CDIM, c = t % CDIM;
    const float* xb = X + (long)b * npg * CDIM + c;
    float s = 0.f;
    for (int n = 0; n < npg; ++n) s += xb[(long)n * CDIM];
    P[t] = s / (float)npg;
}

// ---------------- FC head (tiny: 128x144) -------------------------------------

__global__ __launch_bounds__(256) void fc_kernel(const float* __restrict__ xin,
                                                 const float* __restrict__ W,
                                                 const float* __restrict__ bias,
                                                 float* __restrict__ xout) {
    int t = blockIdx.x * 256 + threadIdx.x;        // 128*144 threads
    int b = t / CDIM, co = t % CDIM;
    const float* xr = xin + b * CDIM;
    float acc = bias[co] + xr[co];                 // bias + residual
    for (int k = 0; k < CDIM; ++k)
        acc += silu_f(xr[k]) * W[k * CDIM + co];
    xout[t] = acc;
}

__global__ __launch_bounds__(256) void out_kernel(const float* __restrict__ xin,
                                                  const float* __restrict__ oW,
                                                  const float* __restrict__ ob,
                                                  float* __restrict__ out) {
    int t = threadIdx.x;                           // 256 = 128*2
    int b = t >> 1, j = t & 1;
    const float* xr = xin + b * CDIM;
    float acc = ob[j];
    for (int k = 0; k < CDIM; ++k)
        acc += silu_f(xr[k]) * oW[k * 2 + j];
    out[t] = fabsf(acc) + 0.0001f;
}

// -----------------------------------------------------------------------------

extern "C" void kernel_launch(void* const* d_in, const int* in_sizes, int n_in,
                              void* d_out, int out_size, void* d_ws, size_t ws_size,
                              hipStream_t stream) {
    const float* features = (const float*)d_in[0];
    const int*   edge     = (const int*)d_in[1];   // [2, E] : row0 = src, row1 = dst
    const float* embed_W  = (const float*)d_in[3];
    const float* embed_b  = (const float*)d_in[4];
    const float* gcn_W    = (const float*)d_in[5]; // [8,144,144]
    const float* gcn_b    = (const float*)d_in[6]; // [8,144]
    const float* fc_W     = (const float*)d_in[7]; // [3,144,144]
    const float* fc_b     = (const float*)d_in[8]; // [3,144]
    const float* out_W    = (const float*)d_in[9]; // [144,2]
    const float* out_b    = (const float*)d_in[10];

    const int  N   = in_sizes[0];                  // 524288
    const int  E   = in_sizes[1] / 2;              // 2097152
    const int  npg = N / BATCH_G;                  // 4096
    const long NC  = (long)N * CDIM;

    const int* esrc = edge;
    const int* edst = edge + E;

    // workspace layout (floats)
    float* ws    = (float*)d_ws;
    float* dinv  = ws;                             // N          (deg -> dinv in place)
    float* norm  = dinv + N;                       // E
    float* X     = norm + E;                       // N*144
    float* L     = X + NC;                         // N*144
    float* PA    = L + NC;                         // 128*144
    float* PB    = PA + BATCH_G * CDIM;            // 128*144

    const int nbN   = (N + 255) / 256;
    const int nbE   = (E + 255) / 256;
    const int nbNC  = (int)((NC + 255) / 256);
    const long escat = (long)E * 36;
    const int nbES  = (int)((escat + 255) / 256);
    const int nbHD  = (BATCH_G * CDIM) / 256;      // 72

    // --- graph normalization (layer-invariant) ---
    deg_init_kernel<<<nbN, 256, 0, stream>>>(dinv, N);
    deg_count_kernel<<<nbE, 256, 0, stream>>>(edst, dinv, E);
    dinv_kernel<<<nbN, 256, 0, stream>>>(dinv, N);
    norm_kernel<<<nbE, 256, 0, stream>>>(esrc, edst, dinv, norm, E);

    // --- embed ---
    embed_kernel<<<nbNC, 256, 0, stream>>>(features, embed_W, embed_b, X, (int)NC);

    // --- 8 GCN layers: fused (WMMA GEMM + residual/bias/self-loop) -> scatter ---
    for (int i = 0; i < KEDGE; ++i) {
        const float* Wi = gcn_W + (long)i * CDIM * CDIM;
        const float* bi = gcn_b + i * CDIM;
        gemm_silu_wmma<<<N / 128, 256, 0, stream>>>(X, Wi, L, dinv, bi);
        edge_scatter_kernel<<<nbES, 256, 0, stream>>>(X, L, esrc, edst, norm, escat);
    }

    // --- pool + FC head ---
    pool_kernel<<<nbHD, 256, 0, stream>>>(X, PA, npg);
    fc_kernel<<<nbHD, 256, 0, stream>>>(PA, fc_W + 0 * CDIM * CDIM, fc_b + 0 * CDIM, PB);
    fc_kernel<<<nbHD, 256, 0, stream>>>(PB, fc_W + 1 * CDIM * CDIM, fc_b + 1 * CDIM, PA);
    fc_kernel<<<nbHD, 256, 0, stream>>>(PA, fc_W + 2 * CDIM * CDIM, fc_b + 2 * CDIM, PB);
    out_kernel<<<1, 256, 0, stream>>>(PB, out_W, out_b, (float*)d_out);
}